// Encoder_83992380441041
// MI455X (gfx1250) — compile-verified
//
#include <hip/hip_runtime.h>

#define D_DIM 128
#define F_DIM 64
#define KSTEPS (D_DIM / 4)   // 32 k-steps of 4 for the D-dim GEMMs

typedef float v2f __attribute__((ext_vector_type(2)));
typedef float v8f __attribute__((ext_vector_type(8)));

__device__ __forceinline__ float wave_sum32(float v) {
#pragma unroll
  for (int off = 16; off >= 1; off >>= 1)
    v += __shfl_xor(v, off, 32);
  return v;
}

// phi(a) = elu(a) + 1 = (a > 0) ? a + 1 : exp(a)
__device__ __forceinline__ float phi(float a) {
  return a > 0.0f ? a + 1.0f : __expf(a);
}

// One block per batch. 256 threads = 8 waves (wave32).
// LDS: yn[T][128] | K[T][64] | V[T][64] | S[64][64] | q[64] | Z[64]
__global__ void linattn_last_kernel(const float* __restrict__ x,
                                    const float* __restrict__ gamma,
                                    const float* __restrict__ beta,
                                    const float* __restrict__ Wk,
                                    const float* __restrict__ Wq,
                                    const float* __restrict__ Wv,
                                    float* __restrict__ out,
                                    int T) {
  extern __shared__ float smem[];
  float* yn = smem;                              // T * 128
  float* sK = yn + (size_t)T * D_DIM;            // T * 64
  float* sV = sK + (size_t)T * F_DIM;            // T * 64
  float* sS = sV + (size_t)T * F_DIM;            // 64 * 64
  float* sq = sS + F_DIM * F_DIM;                // 64
  float* sZ = sq + F_DIM;                        // 64

  const int b    = blockIdx.x;
  const int tid  = threadIdx.x;
  const int lane = tid & 31;
  const int wave = tid >> 5;       // 0..7
  const int half = lane >> 4;      // 0 | 1 (WMMA lane-half)
  const int l16  = lane & 15;

  const float* xb = x + (size_t)b * T * D_DIM;

  // ---------------- Phase 1: LayerNorm, one wave per row ----------------
  const float4 g  = *(const float4*)(gamma + lane * 4);
  const float4 be = *(const float4*)(beta + lane * 4);
  for (int r = wave; r < T; r += 8) {
    const float4 v = *(const float4*)(xb + (size_t)r * D_DIM + lane * 4);
    float mean = wave_sum32(v.x + v.y + v.z + v.w) * (1.0f / D_DIM);
    float dx = v.x - mean, dy = v.y - mean, dz = v.z - mean, dw = v.w - mean;
    float var = wave_sum32(dx * dx + dy * dy + dz * dz + dw * dw) * (1.0f / D_DIM);
    float rstd = rsqrtf(var + 1e-5f);
    float* dst = yn + (size_t)r * D_DIM + lane * 4;
    dst[0] = dx * rstd * g.x + be.x;
    dst[1] = dy * rstd * g.y + be.y;
    dst[2] = dz * rstd * g.z + be.z;
    dst[3] = dw * rstd * g.w + be.w;
  }

  // ------- Phase 2: K = phi(yn*Wk^T), V = yn*Wv^T  (WMMA f32 16x16x4) -------
  // 8 column-jobs (K/V x 4 N-tiles) == 8 waves: each wave owns one weight
  // column, loads its 32 B-fragments from global ONCE, reuses them across
  // all T/16 row tiles. Inner loop: batched LDS A-fragment preload, then a
  // back-to-back WMMA burst (accumulator chaining, no per-op full waits).
  const bool  isV = (wave >= 4);
  const int   nt  = wave & 3;
  const float* W  = isV ? Wv : Wk;
  const int   n   = nt * 16 + l16;            // B/D column for this lane

  v2f bfrag[KSTEPS];
#pragma unroll
  for (int kk = 0; kk < KSTEPS; ++kk)         // B[k][n] = W[n][k]
    bfrag[kk] = *(const v2f*)(W + (size_t)n * D_DIM + kk * 4 + 2 * half);

  __syncthreads();                            // yn ready (B loads overlapped)

  const int MT = T >> 4;
  float* dstKV = isV ? sV : sK;
  for (int mt = 0; mt < MT; ++mt) {
    const int m = mt * 16 + l16;              // A-matrix row for this lane
    v2f afrag[KSTEPS];
#pragma unroll
    for (int kk = 0; kk < KSTEPS; ++kk)
      afrag[kk] = *(const v2f*)(yn + (size_t)m * D_DIM + kk * 4 + 2 * half);
    v8f acc = {};
#pragma unroll
    for (int kk = 0; kk < KSTEPS; ++kk)
      acc = __builtin_amdgcn_wmma_f32_16x16x4_f32(false, afrag[kk],
                                                  false, bfrag[kk],
                                                  (short)0, acc, false, false);
#pragma unroll
    for (int r = 0; r < 8; ++r) {
      const int row = mt * 16 + r + 8 * half; // C/D layout: M = r + 8*half
      float val = acc[r];
      dstKV[(size_t)row * F_DIM + n] = isV ? val : phi(val);
    }
  }
  __syncthreads();

  // ---------------- Phase 3: S = K^T * V  (64x64, K-dim = T) ----------------
  // 16 output tiles over 8 waves; K-dim processed in chunks of 16 k-steps:
  // preload 16 A + 16 B fragments from LDS, then a 16-deep WMMA burst.
  for (int t = wave; t < 16; t += 8) {
    const int mt = t >> 2;
    const int ntt = t & 3;
    const int m = mt * 16 + l16;
    const int nn = ntt * 16 + l16;
    v8f acc = {};
    for (int c = 0; c < T; c += 64) {         // 64 rows = 16 k-steps per chunk
      v2f af[16], bf[16];
#pragma unroll
      for (int j = 0; j < 16; ++j) {
        const int k0 = c + j * 4 + 2 * half;
        af[j].x = sK[(size_t)k0 * F_DIM + m];        // A = K^T
        af[j].y = sK[(size_t)(k0 + 1) * F_DIM + m];
        bf[j].x = sV[(size_t)k0 * F_DIM + nn];
        bf[j].y = sV[(size_t)(k0 + 1) * F_DIM + nn];
      }
#pragma unroll
      for (int j = 0; j < 16; ++j)
        acc = __builtin_amdgcn_wmma_f32_16x16x4_f32(false, af[j], false, bf[j],
                                                    (short)0, acc, false, false);
    }
#pragma unroll
    for (int r = 0; r < 8; ++r)
      sS[(size_t)(mt * 16 + r + 8 * half) * F_DIM + nn] = acc[r];
  }

  // q = phi(yn[T-1] * Wq^T) (only the last row's query matters)
  if (tid < F_DIM) {
    const float* yl = yn + (size_t)(T - 1) * D_DIM;
    const float* wq = Wq + (size_t)tid * D_DIM;
    float s = 0.0f;
#pragma unroll 8
    for (int d = 0; d < D_DIM; ++d) s += yl[d] * wq[d];
    sq[tid] = phi(s);
  } else if (tid < 2 * F_DIM) {
    // Z = column sums of K
    const int f = tid - F_DIM;
    float s = 0.0f;
    for (int t2 = 0; t2 < T; ++t2) s += sK[(size_t)t2 * F_DIM + f];
    sZ[f] = s;
  }
  __syncthreads();

  // ---------------- Phase 4: z[b] = (q . S) / (q . Z + eps) ----------------
  if (tid < F_DIM) {
    float den = 1e-5f;
    float num = 0.0f;
#pragma unroll 8
    for (int i = 0; i < F_DIM; ++i) {
      den += sq[i] * sZ[i];
      num += sq[i] * sS[(size_t)i * F_DIM + tid];
    }
    out[(size_t)b * F_DIM + tid] = num / den;
  }
}

extern "C" void kernel_launch(void* const* d_in, const int* in_sizes, int n_in,
                              void* d_out, int out_size, void* d_ws, size_t ws_size,
                              hipStream_t stream) {
  (void)d_ws; (void)ws_size; (void)n_in;
  const float* x     = (const float*)d_in[0];
  // d_in[1] = batch ids: segments are equal-sized and sorted -> layout is
  // exactly x.reshape(B, T, D); the ids are not needed.
  const float* gamma = (const float*)d_in[2];
  const float* beta  = (const float*)d_in[3];
  const float* Wk    = (const float*)d_in[4];
  const float* Wq    = (const float*)d_in[5];
  const float* Wv    = (const float*)d_in[6];
  float* out = (float*)d_out;

  const int B = out_size / F_DIM;                 // 64
  const int N = in_sizes[0] / D_DIM;              // 16384 rows
  const int T = N / B;                            // 256

  const size_t lds = ((size_t)T * D_DIM            // yn
                      + 2 * (size_t)T * F_DIM      // K, V
                      + (size_t)F_DIM * F_DIM      // S
                      + 2 * F_DIM)                 // q, Z
                     * sizeof(float);              // 279 KB < 320 KB WGP LDS

  // Allow >64KB dynamic LDS (CDNA5 WGP has 320KB). Attribute set is not a
  // stream operation, so it is graph-capture safe and idempotent.
  hipFuncSetAttribute((const void*)linattn_last_kernel,
                      hipFuncAttributeMaxDynamicSharedMemorySize, (int)lds);

  linattn_last_kernel<<<B, 256, lds, stream>>>(x, gamma, beta, Wk, Wq, Wv, out, T);
}